// GAT_2L_26302379720715
// MI455X (gfx1250) — compile-verified
//
#include <hip/hip_runtime.h>
#include <math.h>

// ---------------- CDNA5 WMMA types ----------------
// fp32 WMMA: A,B are 16x4 / 4x16 f32 -> 2 VGPRs per lane; C/D 16x16 f32 -> 8 VGPRs.
typedef __attribute__((ext_vector_type(2))) float v2f;
typedef __attribute__((ext_vector_type(8))) float v8f;

#define NEG_SLOPE 0.2f

// float atomic max via int/uint monotonic punning (init must be -inf).
__device__ __forceinline__ void atomicMaxF(float* addr, float v) {
  if (v >= 0.0f) atomicMax((int*)addr, __float_as_int(v));
  else           atomicMin((unsigned int*)addr, __float_as_uint(v));
}

__global__ void __launch_bounds__(256)
fill_kernel(float* __restrict__ p, int n, float v) {
  int i = blockIdx.x * blockDim.x + threadIdx.x;
  if (i < n) p[i] = v;
}

// ---------------------------------------------------------------------------
// GEMM1: xl1 = x @ W1l, xr1 = x @ W1r.   [N,128] x [128,64] -> [N,64]
// Exact fp32 via V_WMMA_F32_16X16X4_F32, K=128 as 32 chained WMMAs.
// A-frag layout (ISA 7.12.2, 32-bit A 16x4): lane L holds row L%16,
// K = 2*(L/16)+{0,1}  -> one contiguous float2 load per K-step.
// B symmetric with M->N. grid.x = mtiles, block = 128 (wave w = ntile w).
// ---------------------------------------------------------------------------
__global__ void __launch_bounds__(128)
gemm1_kernel(const float* __restrict__ x, const float* __restrict__ W1l,
             const float* __restrict__ W1r, float* __restrict__ xl1,
             float* __restrict__ xr1, int n_nodes) {
  const int lane = threadIdx.x & 31;
  const int wave = threadIdx.x >> 5;          // ntile 0..3
  const int m0   = blockIdx.x * 16;
  const int n0   = wave * 16;
  const int half = lane >> 4;
  const int rc   = lane & 15;                 // A row / B,D col within tile

  int arow = m0 + rc; if (arow >= n_nodes) arow = n_nodes - 1;
  const v2f*   xrow = (const v2f*)(x + (size_t)arow * 128);
  const float* Bl   = W1l + n0 + rc;          // column rc of W1l, stride 64
  const float* Br   = W1r + n0 + rc;

  v8f accL = {}, accR = {};
  #pragma unroll
  for (int kc = 0; kc < 32; ++kc) {
    v2f a = xrow[2 * kc + half];              // K = 4*kc + 2*half + {0,1}
    const int k0 = 4 * kc + 2 * half;
    v2f bl, br;
    bl[0] = Bl[(size_t)(k0 + 0) * 64];
    bl[1] = Bl[(size_t)(k0 + 1) * 64];
    br[0] = Br[(size_t)(k0 + 0) * 64];
    br[1] = Br[(size_t)(k0 + 1) * 64];
    accL = __builtin_amdgcn_wmma_f32_16x16x4_f32(
        false, a, false, bl, (short)0, accL, false, false);
    accR = __builtin_amdgcn_wmma_f32_16x16x4_f32(
        false, a, false, br, (short)0, accR, false, false);
  }

  #pragma unroll
  for (int r = 0; r < 8; ++r) {
    int row = m0 + r + 8 * half;              // C/D layout: lanes 0-15 M=0..7
    if (row < n_nodes) {
      xl1[(size_t)row * 64 + n0 + rc] = accL[r];
      xr1[(size_t)row * 64 + n0 + rc] = accR[r];
    }
  }
}

// ---------------------------------------------------------------------------
// GEMM2: xl2 = h @ W2l, xr2 = h @ W2r.  [N,64] x [64,10] -> [N,16-padded]
// block = 128 (4 waves), each wave owns one M-tile; N padded 10 -> 16.
// ---------------------------------------------------------------------------
__global__ void __launch_bounds__(128)
gemm2_kernel(const float* __restrict__ h, const float* __restrict__ W2l,
             const float* __restrict__ W2r, float* __restrict__ xl2,
             float* __restrict__ xr2, int n_nodes, int mtiles) {
  const int lane  = threadIdx.x & 31;
  const int mtile = blockIdx.x * 4 + (threadIdx.x >> 5);
  if (mtile >= mtiles) return;                // wave-uniform: EXEC stays full
  const int m0   = mtile * 16;
  const int half = lane >> 4;
  const int rc   = lane & 15;

  int arow = m0 + rc; if (arow >= n_nodes) arow = n_nodes - 1;
  const v2f* hrow = (const v2f*)(h + (size_t)arow * 64);
  const bool colok = (rc < 10);

  v8f accL = {}, accR = {};
  #pragma unroll
  for (int kc = 0; kc < 16; ++kc) {
    v2f a = hrow[2 * kc + half];
    const int k0 = 4 * kc + 2 * half;
    v2f bl, br;
    bl[0] = colok ? W2l[(size_t)(k0 + 0) * 10 + rc] : 0.0f;
    bl[1] = colok ? W2l[(size_t)(k0 + 1) * 10 + rc] : 0.0f;
    br[0] = colok ? W2r[(size_t)(k0 + 0) * 10 + rc] : 0.0f;
    br[1] = colok ? W2r[(size_t)(k0 + 1) * 10 + rc] : 0.0f;
    accL = __builtin_amdgcn_wmma_f32_16x16x4_f32(
        false, a, false, bl, (short)0, accL, false, false);
    accR = __builtin_amdgcn_wmma_f32_16x16x4_f32(
        false, a, false, br, (short)0, accR, false, false);
  }

  #pragma unroll
  for (int r = 0; r < 8; ++r) {
    int row = m0 + r + 8 * half;
    if (row < n_nodes) {
      xl2[(size_t)row * 16 + rc] = accL[r];   // cols 10..15 = 0 (B padded)
      xr2[(size_t)row * 16 + rc] = accR[r];
    }
  }
}

// ---------------------------------------------------------------------------
// Layer-1 edge pass 1: logits[e,h] = att1[h,:] . lrelu(xl1[src]+xr1[dst])
// 16 lanes per edge (4 channels each); segment-max via float atomic max.
// ---------------------------------------------------------------------------
__global__ void __launch_bounds__(256)
l1_edge_logits(const int* __restrict__ ei, const float* __restrict__ xl1,
               const float* __restrict__ xr1, const float* __restrict__ att1,
               float* __restrict__ logits1, float* __restrict__ m1, int E) {
  int gid = blockIdx.x * blockDim.x + threadIdx.x;
  int g   = gid >> 4;
  bool act = (g < E);
  if (g >= E) g = E - 1;                      // keep lanes alive for shfl
  const int l   = threadIdx.x & 15;
  const int src = ei[g];
  const int dst = ei[E + g];

  const float4 xv = *(const float4*)(xl1 + (size_t)src * 64 + l * 4);
  const float4 rv = *(const float4*)(xr1 + (size_t)dst * 64 + l * 4);
  const float4 at = *(const float4*)(att1 + l * 4);

  float p = 0.0f, s;
  s = xv.x + rv.x; s = (s > 0.f) ? s : NEG_SLOPE * s; p += s * at.x;
  s = xv.y + rv.y; s = (s > 0.f) ? s : NEG_SLOPE * s; p += s * at.y;
  s = xv.z + rv.z; s = (s > 0.f) ? s : NEG_SLOPE * s; p += s * at.z;
  s = xv.w + rv.w; s = (s > 0.f) ? s : NEG_SLOPE * s; p += s * at.w;

  p += __shfl_xor(p, 1, 32);                  // reduce 4 lanes -> one head
  p += __shfl_xor(p, 2, 32);

  if (act && (l & 3) == 0) {
    int hd = l >> 2;
    logits1[(size_t)g * 4 + hd] = p;
    atomicMaxF(m1 + (size_t)dst * 4 + hd, p);
  }
}

// Layer-1 edge pass 2: w = exp(logit - m[dst]); denom += w; acc += w*xl1[src]
__global__ void __launch_bounds__(256)
l1_edge_aggregate(const int* __restrict__ ei, const float* __restrict__ xl1,
                  const float* __restrict__ logits1, const float* __restrict__ m1,
                  float* __restrict__ den1, float* __restrict__ hacc, int E) {
  int g = (blockIdx.x * blockDim.x + threadIdx.x) >> 4;
  if (g >= E) return;
  const int l   = threadIdx.x & 15;
  const int src = ei[g];
  const int dst = ei[E + g];
  const int hd  = l >> 2;

  float w = __expf(logits1[(size_t)g * 4 + hd] - m1[(size_t)dst * 4 + hd]);
  if ((l & 3) == 0) atomicAdd(den1 + (size_t)dst * 4 + hd, w);

  const float4 v = *(const float4*)(xl1 + (size_t)src * 64 + l * 4);
  float* o = hacc + (size_t)dst * 64 + l * 4;
  atomicAdd(o + 0, w * v.x);
  atomicAdd(o + 1, w * v.y);
  atomicAdd(o + 2, w * v.z);
  atomicAdd(o + 3, w * v.w);
}

// Layer-1 finish: h = elu(acc/denom + b1)   (in place on hacc)
__global__ void __launch_bounds__(256)
l1_finish(float* __restrict__ hacc, const float* __restrict__ den1,
          const float* __restrict__ b1, int total) {
  int i = blockIdx.x * blockDim.x + threadIdx.x;
  if (i >= total) return;
  int n = i >> 6, ch = i & 63, hd = ch >> 4;
  float d = den1[(size_t)n * 4 + hd];
  float v = (d > 0.0f) ? hacc[i] / d : 0.0f;  // zero in-degree -> segment_sum = 0
  v += b1[ch];
  hacc[i] = (v > 0.0f) ? v : expm1f(v);
}

// Layer-2 edge pass 1: 4 lanes per edge over the padded 16-wide features.
__global__ void __launch_bounds__(256)
l2_edge_logits(const int* __restrict__ ei, const float* __restrict__ xl2,
               const float* __restrict__ xr2, const float* __restrict__ att2,
               float* __restrict__ logits2, float* __restrict__ m2, int E) {
  int gid = blockIdx.x * blockDim.x + threadIdx.x;
  int g   = gid >> 2;
  bool act = (g < E);
  if (g >= E) g = E - 1;
  const int l   = threadIdx.x & 3;
  const int src = ei[g];
  const int dst = ei[E + g];
  const int c0  = l * 4;

  const float4 a = *(const float4*)(xl2 + (size_t)src * 16 + c0);
  const float4 b = *(const float4*)(xr2 + (size_t)dst * 16 + c0);

  float p = 0.0f, s;
  s = a.x + b.x; s = (s > 0.f) ? s : NEG_SLOPE * s; if (c0 + 0 < 10) p += s * att2[c0 + 0];
  s = a.y + b.y; s = (s > 0.f) ? s : NEG_SLOPE * s; if (c0 + 1 < 10) p += s * att2[c0 + 1];
  s = a.z + b.z; s = (s > 0.f) ? s : NEG_SLOPE * s; if (c0 + 2 < 10) p += s * att2[c0 + 2];
  s = a.w + b.w; s = (s > 0.f) ? s : NEG_SLOPE * s; if (c0 + 3 < 10) p += s * att2[c0 + 3];

  p += __shfl_xor(p, 1, 32);
  p += __shfl_xor(p, 2, 32);

  if (act && l == 0) {
    logits2[g] = p;
    atomicMaxF(m2 + dst, p);
  }
}

__global__ void __launch_bounds__(256)
l2_edge_aggregate(const int* __restrict__ ei, const float* __restrict__ xl2,
                  const float* __restrict__ logits2, const float* __restrict__ m2,
                  float* __restrict__ den2, float* __restrict__ acc2, int E) {
  int g = (blockIdx.x * blockDim.x + threadIdx.x) >> 2;
  if (g >= E) return;
  const int l   = threadIdx.x & 3;
  const int src = ei[g];
  const int dst = ei[E + g];
  const int c0  = l * 4;

  float w = __expf(logits2[g] - m2[dst]);
  if (l == 0) atomicAdd(den2 + dst, w);

  const float4 v = *(const float4*)(xl2 + (size_t)src * 16 + c0);
  float* o = acc2 + (size_t)dst * 16 + c0;
  if (c0 + 0 < 10) atomicAdd(o + 0, w * v.x);
  if (c0 + 1 < 10) atomicAdd(o + 1, w * v.y);
  if (c0 + 2 < 10) atomicAdd(o + 2, w * v.z);
  if (c0 + 3 < 10) atomicAdd(o + 3, w * v.w);
}

// Layer-2 finish (H2=1, mean over heads is identity): out = acc/denom + b2
__global__ void __launch_bounds__(256)
l2_finish(const float* __restrict__ acc2, const float* __restrict__ den2,
          const float* __restrict__ b2, float* __restrict__ out, int total) {
  int i = blockIdx.x * blockDim.x + threadIdx.x;
  if (i >= total) return;
  int n = i / 10, c = i % 10;
  float d = den2[n];
  float v = (d > 0.0f) ? acc2[(size_t)n * 16 + c] / d : 0.0f;
  out[i] = v + b2[c];
}

// ---------------------------------------------------------------------------
extern "C" void kernel_launch(void* const* d_in, const int* in_sizes, int n_in,
                              void* d_out, int out_size, void* d_ws, size_t ws_size,
                              hipStream_t stream) {
  const float* x    = (const float*)d_in[0];
  const int*   ei   = (const int*)d_in[1];      // edge_index (int32 on device)
  const float* W1l  = (const float*)d_in[2];
  const float* W1r  = (const float*)d_in[3];
  const float* att1 = (const float*)d_in[4];
  const float* b1   = (const float*)d_in[5];
  const float* W2l  = (const float*)d_in[6];
  const float* W2r  = (const float*)d_in[7];
  const float* att2 = (const float*)d_in[8];
  const float* b2   = (const float*)d_in[9];
  float* out = (float*)d_out;

  const int N = in_sizes[0] / 128;
  const int E = in_sizes[1] / 2;

  // Workspace layout (floats), ~66 MB total
  float* ws = (float*)d_ws;
  size_t o = 0;
  float* xl1  = ws + o; o += (size_t)N * 64;
  float* xr1  = ws + o; o += (size_t)N * 64;
  float* hacc = ws + o; o += (size_t)N * 64;   // layer-1 accumulator, then h1
  float* lg1  = ws + o; o += (size_t)E * 4;
  float* m1   = ws + o; o += (size_t)N * 4;
  float* dn1  = ws + o; o += (size_t)N * 4;
  float* xl2  = ws + o; o += (size_t)N * 16;
  float* xr2  = ws + o; o += (size_t)N * 16;
  float* acc2 = ws + o; o += (size_t)N * 16;
  float* lg2  = ws + o; o += (size_t)E;
  float* m2   = ws + o; o += (size_t)N;
  float* dn2  = ws + o; o += (size_t)N;

  auto blocks = [](long long n) { return (unsigned)((n + 255) / 256); };
  const float NEG_INF = -__builtin_inff();

  // Re-initialize accumulators/maxima every launch (graph-replay safe).
  fill_kernel<<<blocks((long long)N * 64), 256, 0, stream>>>(hacc, N * 64, 0.0f);
  fill_kernel<<<blocks((long long)N * 4), 256, 0, stream>>>(m1, N * 4, NEG_INF);
  fill_kernel<<<blocks((long long)N * 4), 256, 0, stream>>>(dn1, N * 4, 0.0f);
  fill_kernel<<<blocks((long long)N * 16), 256, 0, stream>>>(acc2, N * 16, 0.0f);
  fill_kernel<<<blocks(N), 256, 0, stream>>>(m2, N, NEG_INF);
  fill_kernel<<<blocks(N), 256, 0, stream>>>(dn2, N, 0.0f);

  const int mtiles = (N + 15) / 16;

  // ---- Layer 1 ----
  gemm1_kernel<<<mtiles, 128, 0, stream>>>(x, W1l, W1r, xl1, xr1, N);
  l1_edge_logits<<<blocks((long long)E * 16), 256, 0, stream>>>(ei, xl1, xr1, att1,
                                                                lg1, m1, E);
  l1_edge_aggregate<<<blocks((long long)E * 16), 256, 0, stream>>>(ei, xl1, lg1, m1,
                                                                   dn1, hacc, E);
  l1_finish<<<blocks((long long)N * 64), 256, 0, stream>>>(hacc, dn1, b1, N * 64);

  // ---- Layer 2 ----
  gemm2_kernel<<<(mtiles + 3) / 4, 128, 0, stream>>>(hacc, W2l, W2r, xl2, xr2, N, mtiles);
  l2_edge_logits<<<blocks((long long)E * 4), 256, 0, stream>>>(ei, xl2, xr2, att2,
                                                               lg2, m2, E);
  l2_edge_aggregate<<<blocks((long long)E * 4), 256, 0, stream>>>(ei, xl2, lg2, m2,
                                                                  dn2, acc2, E);
  l2_finish<<<blocks((long long)N * 10), 256, 0, stream>>>(acc2, dn2, b2, out, N * 10);
}